// MVN_DDI_Block_15375982920242
// MI455X (gfx1250) — compile-verified
//
#include <hip/hip_runtime.h>
#include <math.h>

// ---------------------------------------------------------------------------
// Graph Transformer layer for MI455X (gfx1250, wave32).
//   Q/K/V = h @ W{Q,K,V}; proj_e = e @ WE        -> v_wmma_f32_16x16x4_f32
//   score = K[src]*Q[dst]*0.25*proj_e            -> fused with edge GEMM
//   s = exp(clamp(sum score))                    -> scatter via f32 atomics
//   h_out = segsum(V[src]*s)/(segsum(s)+1e-6)
//
// Data movement: tiles staged into LDS with GLOBAL_LOAD_ASYNC_TO_LDS_B128
// (ASYNCcnt path, no VGPR round-trip), e/e_out streamed non-temporally so
// the node tensors (Q/K/V/wV/z ~104MB) stay resident in the 192MB L2.
// ---------------------------------------------------------------------------

typedef __attribute__((ext_vector_type(2))) float v2f;
typedef __attribute__((ext_vector_type(4))) float f4;
typedef __attribute__((ext_vector_type(8))) float v8f;

#define TILE_ROWS 64
#define LDS_PITCH 132   // 128 + 4 floats pad: column reads hit distinct banks

__device__ __forceinline__ v8f wmma4(v2f a, v2f b, v8f c) {
  // D = A(16x4,f32) * B(4x16,f32) + C(16x16,f32)
  return __builtin_amdgcn_wmma_f32_16x16x4_f32(
      /*neg_a=*/false, a, /*neg_b=*/false, b,
      /*c_mod=*/(short)0, c, /*reuse_a=*/false, /*reuse_b=*/false);
}

__device__ __forceinline__ void atomic_fadd(float* p, float v) {
  __hip_atomic_fetch_add(p, v, __ATOMIC_RELAXED, __HIP_MEMORY_SCOPE_AGENT);
}

// Async 16B global -> LDS copy (per lane). vdst VGPR = LDS byte address;
// flat pointer's low 32 bits are the LDS byte address per the aperture rule.
// th:TH_LOAD_NT: streamed data, don't evict L2-resident node tensors.
__device__ __forceinline__ void async_ld_b128_nt(const void* gptr, void* lptr) {
  unsigned lds = (unsigned)(uintptr_t)lptr;
  asm volatile("global_load_async_to_lds_b128 %0, %1, off th:TH_LOAD_NT"
               :: "v"(lds), "v"(gptr) : "memory");
}
__device__ __forceinline__ void wait_async0() {
  asm volatile("s_wait_asynccnt 0" ::: "memory");
}

// ---------------------------------------------------------------------------
__global__ void zero_kernel(float* __restrict__ p, int n) {
  int i = blockIdx.x * blockDim.x + threadIdx.x;
  if (i < n) p[i] = 0.0f;
}

// ---------------------------------------------------------------------------
// Fused Q/K/V projection: one 64x128 tile of h per workgroup, 8 waves each
// owning one 16-wide column tile; A fragment (from LDS) reused across
// 4 row-tiles x 3 weight matrices = 12 WMMAs per K-step.
__global__ __launch_bounds__(256) void qkv_kernel(
    const float* __restrict__ h,
    const float* __restrict__ WQ, const float* __restrict__ WK,
    const float* __restrict__ WV,
    float* __restrict__ Q, float* __restrict__ K, float* __restrict__ V,
    int N) {
  __shared__ float hs[TILE_ROWS * LDS_PITCH];
  const int tid = threadIdx.x;
  const int row0 = blockIdx.x * TILE_ROWS;

  for (int i = tid; i < TILE_ROWS * 32; i += 256) {  // 32 x f4 per row
    int r = i >> 5, c4 = (i & 31) << 2;
    int gr = row0 + r;
    float* lp = hs + r * LDS_PITCH + c4;
    if (gr < N) {
      async_ld_b128_nt(h + (size_t)gr * 128 + c4, lp);
    } else {
      f4 zero = {0.f, 0.f, 0.f, 0.f};
      *(f4*)lp = zero;
    }
  }
  wait_async0();
  __syncthreads();

  const int wave = tid >> 5, lane = tid & 31;
  const int m  = lane & 15;          // matrix row (A) / col (B,C)
  const int kk = (lane >> 4) << 1;   // upper lane-half holds K=2,3
  const int col = wave * 16 + m;     // this wave's output column

  v8f aq[4] = {{0}}; v8f ak[4] = {{0}}; v8f av[4] = {{0}};
  for (int k0 = 0; k0 < 128; k0 += 4) {
    const int kr = k0 + kk;
    v2f bq, bk, bv;
    bq[0] = WQ[kr * 128 + col]; bq[1] = WQ[(kr + 1) * 128 + col];
    bk[0] = WK[kr * 128 + col]; bk[1] = WK[(kr + 1) * 128 + col];
    bv[0] = WV[kr * 128 + col]; bv[1] = WV[(kr + 1) * 128 + col];
#pragma unroll
    for (int rt = 0; rt < 4; ++rt) {
      v2f a;
      a[0] = hs[(rt * 16 + m) * LDS_PITCH + kr];
      a[1] = hs[(rt * 16 + m) * LDS_PITCH + kr + 1];
      aq[rt] = wmma4(a, bq, aq[rt]);
      ak[rt] = wmma4(a, bk, ak[rt]);
      av[rt] = wmma4(a, bv, av[rt]);
    }
  }

  const int rhalf = (lane >> 4) << 3;  // lanes 16-31 hold rows M+8
  if (row0 + TILE_ROWS <= N) {
    // Fast path: full tile, no per-element predication.
#pragma unroll
    for (int rt = 0; rt < 4; ++rt) {
#pragma unroll
      for (int v = 0; v < 8; ++v) {
        size_t o = (size_t)(row0 + rt * 16 + rhalf + v) * 128 + col;
        Q[o] = aq[rt][v]; K[o] = ak[rt][v]; V[o] = av[rt][v];
      }
    }
  } else {
#pragma unroll
    for (int rt = 0; rt < 4; ++rt) {
#pragma unroll
      for (int v = 0; v < 8; ++v) {
        int r = row0 + rt * 16 + rhalf + v;
        if (r < N) {
          size_t o = (size_t)r * 128 + col;
          Q[o] = aq[rt][v]; K[o] = ak[rt][v]; V[o] = av[rt][v];
        }
      }
    }
  }
}

// ---------------------------------------------------------------------------
// Edge kernel: proj_e GEMM (WMMA) -> LDS, then per-(edge,head) score,
// e_out (non-temporal), and scatter into wV / z with f32 atomics.
__global__ __launch_bounds__(256) void edge_kernel(
    const float* __restrict__ e, const int* __restrict__ eidx,
    const float* __restrict__ WE,
    const float* __restrict__ Q, const float* __restrict__ K,
    const float* __restrict__ V,
    float* __restrict__ e_out, float* __restrict__ wV, float* __restrict__ z,
    int N, int E) {
  __shared__ float sbuf[TILE_ROWS * LDS_PITCH];  // e tile, then proj_e tile
  const int tid = threadIdx.x;
  const int e0 = blockIdx.x * TILE_ROWS;

  // Stage e tile straight into LDS (async, non-temporal).
  for (int i = tid; i < TILE_ROWS * 32; i += 256) {
    int r = i >> 5, c4 = (i & 31) << 2;
    int ge = e0 + r;
    float* lp = sbuf + r * LDS_PITCH + c4;
    if (ge < E) {
      async_ld_b128_nt(e + (size_t)ge * 128 + c4, lp);
    } else {
      f4 zero = {0.f, 0.f, 0.f, 0.f};
      *(f4*)lp = zero;
    }
  }
  wait_async0();
  __syncthreads();

  const int wave = tid >> 5, lane = tid & 31;
  const int m  = lane & 15;
  const int kk = (lane >> 4) << 1;
  const int col = wave * 16 + m;

  v8f acc[4] = {{0}};
  for (int k0 = 0; k0 < 128; k0 += 4) {
    const int kr = k0 + kk;
    v2f b;
    b[0] = WE[kr * 128 + col]; b[1] = WE[(kr + 1) * 128 + col];
#pragma unroll
    for (int rt = 0; rt < 4; ++rt) {
      v2f a;
      a[0] = sbuf[(rt * 16 + m) * LDS_PITCH + kr];
      a[1] = sbuf[(rt * 16 + m) * LDS_PITCH + kr + 1];
      acc[rt] = wmma4(a, b, acc[rt]);
    }
  }
  __syncthreads();  // all waves done reading the e tile

  // Reuse sbuf as proj_e tile (64 edges x 128 feats)
  const int rhalf = (lane >> 4) << 3;
#pragma unroll
  for (int rt = 0; rt < 4; ++rt)
#pragma unroll
    for (int v = 0; v < 8; ++v)
      sbuf[(rt * 16 + rhalf + v) * LDS_PITCH + col] = acc[rt][v];
  __syncthreads();

  // Phase 2: one (edge, head) per task; 64*8 = 512 tasks over 256 threads.
  const float SCALE = 0.25f;  // 1/sqrt(HEAD_DIM=16)
  for (int task = tid; task < TILE_ROWS * 8; task += 256) {
    const int el = task >> 3, hd = task & 7;
    const int eg = e0 + el;
    if (eg >= E) continue;
    const int src = eidx[eg];
    const int dst = eidx[E + eg];

    const float* kp = K + (size_t)src * 128 + hd * 16;
    const float* qp = Q + (size_t)dst * 128 + hd * 16;
    const float* pe = sbuf + el * LDS_PITCH + hd * 16;

    float sum = 0.f;
    f4 sc[4];
#pragma unroll
    for (int d4 = 0; d4 < 4; ++d4) {
      f4 kv = *(const f4*)(kp + d4 * 4);
      f4 qv = *(const f4*)(qp + d4 * 4);
      f4 pv = *(const f4*)(pe + d4 * 4);
      f4 s = kv * qv * pv * SCALE;
      sc[d4] = s;
      sum += s.x + s.y + s.z + s.w;
    }
    float* eo = e_out + (size_t)eg * 128 + hd * 16;
#pragma unroll
    for (int d4 = 0; d4 < 4; ++d4)
      __builtin_nontemporal_store(sc[d4], (f4*)(eo + d4 * 4));

    float sv = __expf(fminf(fmaxf(sum, -5.f), 5.f));
    atomic_fadd(z + (size_t)dst * 8 + hd, sv);

    const float* vp = V + (size_t)src * 128 + hd * 16;
    float* wp = wV + (size_t)dst * 128 + hd * 16;
#pragma unroll
    for (int d4 = 0; d4 < 4; ++d4) {
      f4 vv = *(const f4*)(vp + d4 * 4);
      atomic_fadd(wp + d4 * 4 + 0, vv.x * sv);
      atomic_fadd(wp + d4 * 4 + 1, vv.y * sv);
      atomic_fadd(wp + d4 * 4 + 2, vv.z * sv);
      atomic_fadd(wp + d4 * 4 + 3, vv.w * sv);
    }
  }
}

// ---------------------------------------------------------------------------
__global__ void finalize_kernel(const float* __restrict__ wV,
                                const float* __restrict__ z,
                                float* __restrict__ h_out, int N) {
  int i = blockIdx.x * blockDim.x + threadIdx.x;
  if (i >= N * 128) return;
  int n = i >> 7;
  int hd = (i >> 4) & 7;
  h_out[i] = wV[i] / (z[n * 8 + hd] + 1e-6f);
}

// ---------------------------------------------------------------------------
extern "C" void kernel_launch(void* const* d_in, const int* in_sizes, int n_in,
                              void* d_out, int out_size, void* d_ws, size_t ws_size,
                              hipStream_t stream) {
  const float* h  = (const float*)d_in[0];
  const float* e  = (const float*)d_in[1];
  const int* eidx = (const int*)d_in[2];  // JAX default x64 off -> int32
  const float* WQ = (const float*)d_in[3];
  const float* WK = (const float*)d_in[4];
  const float* WV = (const float*)d_in[5];
  const float* WE = (const float*)d_in[6];

  const int N = in_sizes[0] / 128;
  const int E = in_sizes[1] / 128;

  float* ws = (float*)d_ws;
  const size_t NN = (size_t)N * 128;
  float* Qw  = ws;
  float* Kw  = ws + NN;
  float* Vw  = ws + 2 * NN;
  float* wVa = ws + 3 * NN;
  float* za  = ws + 4 * NN;  // N*8 floats

  float* h_out = (float*)d_out;
  float* e_out = h_out + NN;

  const int zn = (int)(NN + (size_t)N * 8);  // wV and z are contiguous
  zero_kernel<<<(zn + 255) / 256, 256, 0, stream>>>(wVa, zn);

  qkv_kernel<<<(N + TILE_ROWS - 1) / TILE_ROWS, 256, 0, stream>>>(
      h, WQ, WK, WV, Qw, Kw, Vw, N);

  edge_kernel<<<(E + TILE_ROWS - 1) / TILE_ROWS, 256, 0, stream>>>(
      e, eidx, WE, Qw, Kw, Vw, e_out, wVa, za, N, E);

  finalize_kernel<<<((int)NN + 255) / 256, 256, 0, stream>>>(wVa, za, h_out, N);
}